// RecipeGNN_61838939128118
// MI455X (gfx1250) — compile-verified
//
#include <hip/hip_runtime.h>

typedef float v2f __attribute__((ext_vector_type(2)));
typedef float v8f __attribute__((ext_vector_type(8)));

#define K_DIM 128   // inner dim of every GEMM in this net (DIN == DH == 128)

// ---------------------------------------------------------------------------
// GEMM: T[N x DOUT] = A[N x 128] @ W[128 x DOUT], full fp32 via WMMA f32 16x16x4
// Block = 256 threads = 8 wave32. Each wave owns one 16x16 output tile.
// DOUT=128 -> 8 col tiles, 1 row tile/block (16 rows).
// DOUT=64  -> 4 col tiles, 2 row tiles/block (32 rows).
// ---------------------------------------------------------------------------
template <int DOUT>
__global__ __launch_bounds__(256) void gemm_f32_wmma(
    const float* __restrict__ A, const float* __restrict__ W,
    float* __restrict__ T, int N) {
  constexpr int CT = DOUT / 16;    // column tiles covered by the 8 waves
  constexpr int RT = 8 / CT;       // row tiles per block
  constexpr int ROWS = RT * 16;
  __shared__ float As[ROWS][K_DIM + 4];   // +4 pad: avoid 16-way LDS bank conflicts

  const int tid  = threadIdx.x;
  const int wave = tid >> 5;
  const int lane = tid & 31;
  const int rowTile = wave / CT;
  const int colTile = wave % CT;
  const long rowBase = (long)blockIdx.x * ROWS;

  // Cooperative stage of the A row-tile into LDS (coalesced 128B rows)
  for (int i = tid; i < ROWS * K_DIM; i += 256) {
    const int r = i >> 7;        // /128
    const int c = i & 127;
    const long gr = rowBase + r;
    As[r][c] = (gr < N) ? A[gr * K_DIM + c] : 0.0f;
  }
  __syncthreads();

  const int half = lane >> 4;          // lane half selects K pair / M+8
  const int l16  = lane & 15;
  const int n0   = colTile * 16 + l16; // output column this lane owns
  const int arow = rowTile * 16 + l16; // A row this lane feeds

  v8f acc = {};
  #pragma unroll
  for (int kk = 0; kk < K_DIM; kk += 4) {
    const int ka = kk + 2 * half;
    v2f a, b;
    // A 16x4 f32 fragment: lanes 0-15 hold K={kk,kk+1}, lanes 16-31 K={kk+2,kk+3}
    a.x = As[arow][ka];
    a.y = As[arow][ka + 1];
    // B 4x16 f32 fragment (mirror layout), coalesced across lanes in n
    b.x = W[ka * DOUT + n0];
    b.y = W[(ka + 1) * DOUT + n0];
    acc = __builtin_amdgcn_wmma_f32_16x16x4_f32(
        /*neg_a=*/false, a, /*neg_b=*/false, b,
        /*c_mod=*/(short)0, acc, /*reuse_a=*/false, /*reuse_b=*/false);
  }

  // C/D layout: VGPR v, lanes 0-15 -> M=v ; lanes 16-31 -> M=v+8 ; N=lane%16
  const long mBase = rowBase + rowTile * 16 + 8 * half;
  #pragma unroll
  for (int v = 0; v < 8; ++v) {
    const long m = mBase + v;
    if (m < N) T[m * DOUT + n0] = acc[v];
  }
}

// ---------------------------------------------------------------------------
// Degree / normalization
// ---------------------------------------------------------------------------
__global__ void init_deg_kernel(float* deg, int N) {
  const long i = (long)blockIdx.x * blockDim.x + threadIdx.x;
  if (i < N) deg[i] = 1.0f;   // self loop
}

__global__ void count_deg_kernel(const int* __restrict__ dst, float* deg, int E) {
  const long e = (long)blockIdx.x * blockDim.x + threadIdx.x;
  if (e < E) atomicAdd(&deg[dst[e]], 1.0f);
}

__global__ void finalize_dinv_kernel(float* deg, int N) {
  const long i = (long)blockIdx.x * blockDim.x + threadIdx.x;
  if (i < N) deg[i] = rsqrtf(deg[i]);   // deg >= 1 always (self loop)
}

// ---------------------------------------------------------------------------
// Aggregation: out = D^{-1/2}(A+I)D^{-1/2} @ T
// agg_self writes the self-loop term (doubles as output init),
// agg_edges scatters edge messages with f32 atomics (L2-resident).
// ---------------------------------------------------------------------------
template <int D>
__global__ void agg_self_kernel(const float* __restrict__ T,
                                const float* __restrict__ dinv,
                                float* __restrict__ out, int N) {
  const long idx = (long)blockIdx.x * blockDim.x + threadIdx.x;
  if (idx >= (long)N * D) return;
  const long i = idx / D;
  const float w = dinv[i];
  out[idx] = T[idx] * w * w;
}

template <int D>
__global__ void agg_edges_kernel(const float* __restrict__ T,
                                 const int* __restrict__ src,
                                 const int* __restrict__ dst,
                                 const float* __restrict__ dinv,
                                 float* __restrict__ out, int E) {
  constexpr int TPE = D / 4;   // threads per edge, float4 each
  const long idx = (long)blockIdx.x * blockDim.x + threadIdx.x;
  const long e = idx / TPE;
  const int  t = (int)(idx % TPE);
  if (e >= E) return;
  const int s = src[e];
  const int d = dst[e];
  const float norm = dinv[s] * dinv[d];
  const float4 v = ((const float4*)(T + (long)s * D))[t];
  float* op = out + (long)d * D + t * 4;
  atomicAdd(op + 0, v.x * norm);
  atomicAdd(op + 1, v.y * norm);
  atomicAdd(op + 2, v.z * norm);
  atomicAdd(op + 3, v.w * norm);
}

template <int D>
__global__ void bias_relu_kernel(float* __restrict__ h,
                                 const float* __restrict__ b, long total) {
  const long idx = (long)blockIdx.x * blockDim.x + threadIdx.x;
  if (idx >= total) return;
  const int d = (int)(idx % D);
  h[idx] = fmaxf(h[idx] + b[d], 0.0f);
}

// ---------------------------------------------------------------------------
// Global mean pool
// ---------------------------------------------------------------------------
__global__ void zero_kernel(float* p, long n) {
  const long i = (long)blockIdx.x * blockDim.x + threadIdx.x;
  if (i < n) p[i] = 0.0f;
}

__global__ void pool_count_kernel(const int* __restrict__ batch, float* counts, int N) {
  const long i = (long)blockIdx.x * blockDim.x + threadIdx.x;
  if (i < N) atomicAdd(&counts[batch[i]], 1.0f);
}

__global__ void pool_sum_kernel(const float* __restrict__ h,
                                const int* __restrict__ batch,
                                float* sums, int N) {
  const long idx = (long)blockIdx.x * blockDim.x + threadIdx.x;
  if (idx >= (long)N * 64) return;
  const long i = idx / 64;
  const int  d = (int)(idx % 64);
  atomicAdd(&sums[(long)batch[i] * 64 + d], h[idx]);
}

// out = (sums + cnt*b3) / max(cnt,1)  == mean over nodes of (agg + b3), exact
// including the empty-graph case (gives 0).
__global__ void pool_final_kernel(const float* __restrict__ sums,
                                  const float* __restrict__ counts,
                                  const float* __restrict__ b3,
                                  float* __restrict__ out, int G) {
  const long idx = (long)blockIdx.x * blockDim.x + threadIdx.x;
  if (idx >= (long)G * 64) return;
  const long g = idx / 64;
  const int  d = (int)(idx % 64);
  const float c = counts[g];
  out[idx] = (sums[idx] + c * b3[d]) / fmaxf(c, 1.0f);
}

// ---------------------------------------------------------------------------
// Launch
// ---------------------------------------------------------------------------
static inline int cdiv(long a, long b) { return (int)((a + b - 1) / b); }

extern "C" void kernel_launch(void* const* d_in, const int* in_sizes, int n_in,
                              void* d_out, int out_size, void* d_ws, size_t ws_size,
                              hipStream_t stream) {
  const float* x   = (const float*)d_in[0];
  const int*   ei  = (const int*)d_in[1];   // [2, E] flat: src then dst
  const int*   bat = (const int*)d_in[2];
  const float* W1  = (const float*)d_in[3];
  const float* b1  = (const float*)d_in[4];
  const float* W2  = (const float*)d_in[5];
  const float* b2  = (const float*)d_in[6];
  const float* W3  = (const float*)d_in[7];
  const float* b3  = (const float*)d_in[8];
  (void)n_in; (void)ws_size;

  const int N = in_sizes[0] / 128;
  const int E = in_sizes[1] / 2;
  const int G = out_size / 64;
  const int* src = ei;
  const int* dst = ei + E;

  // Workspace layout (floats): dinv | T (N*128) | AGG (N*128) | counts | sums
  float* ws = (float*)d_ws;
  size_t o = 0;
  float* dinv = ws + o;   o += ((size_t)N + 63) & ~(size_t)63;
  float* T    = ws + o;   o += (size_t)N * 128;
  float* AGG  = ws + o;   o += (size_t)N * 128;
  float* counts = ws + o; o += (size_t)G;       // counts + sums contiguous
  float* sums   = ws + o; o += (size_t)G * 64;

  const int B = 256;
  float* out = (float*)d_out;

  // ---- degree normalization -------------------------------------------------
  init_deg_kernel<<<cdiv(N, B), B, 0, stream>>>(dinv, N);
  count_deg_kernel<<<cdiv(E, B), B, 0, stream>>>(dst, dinv, E);
  finalize_dinv_kernel<<<cdiv(N, B), B, 0, stream>>>(dinv, N);

  // ---- layer 1: 128 -> 128 --------------------------------------------------
  gemm_f32_wmma<128><<<cdiv(N, 16), B, 0, stream>>>(x, W1, T, N);
  agg_self_kernel<128><<<cdiv((long)N * 128, B), B, 0, stream>>>(T, dinv, AGG, N);
  agg_edges_kernel<128><<<cdiv((long)E * 32, B), B, 0, stream>>>(T, src, dst, dinv, AGG, E);
  bias_relu_kernel<128><<<cdiv((long)N * 128, B), B, 0, stream>>>(AGG, b1, (long)N * 128);

  // ---- layer 2: 128 -> 128 --------------------------------------------------
  gemm_f32_wmma<128><<<cdiv(N, 16), B, 0, stream>>>(AGG, W2, T, N);
  agg_self_kernel<128><<<cdiv((long)N * 128, B), B, 0, stream>>>(T, dinv, AGG, N);
  agg_edges_kernel<128><<<cdiv((long)E * 32, B), B, 0, stream>>>(T, src, dst, dinv, AGG, E);
  bias_relu_kernel<128><<<cdiv((long)N * 128, B), B, 0, stream>>>(AGG, b2, (long)N * 128);

  // ---- layer 3: 128 -> 64 (bias folded into pool) ---------------------------
  gemm_f32_wmma<64><<<cdiv(N, 32), B, 0, stream>>>(AGG, W3, T, N);
  agg_self_kernel<64><<<cdiv((long)N * 64, B), B, 0, stream>>>(T, dinv, AGG, N);
  agg_edges_kernel<64><<<cdiv((long)E * 16, B), B, 0, stream>>>(T, src, dst, dinv, AGG, E);

  // ---- global mean pool -----------------------------------------------------
  zero_kernel<<<cdiv((long)G * 65, B), B, 0, stream>>>(counts, (long)G * 65);
  pool_count_kernel<<<cdiv(N, B), B, 0, stream>>>(bat, counts, N);
  pool_sum_kernel<<<cdiv((long)N * 64, B), B, 0, stream>>>(AGG, bat, sums, N);
  pool_final_kernel<<<cdiv((long)G * 64, B), B, 0, stream>>>(sums, counts, b3, out, G);
}